// PagedAttentionBlock_63943473103533
// MI455X (gfx1250) — compile-verified
//
#include <hip/hip_runtime.h>

// Problem constants (match reference)
#define NB 16      // batch
#define NH 16      // heads
#define HD 128     // head dim
#define BLK 16     // tokens per cache page
#define BPS 128    // pages per sequence
#define MAXS 2048  // max sequence length
#define GROUPS 128 // 16-key blocks per sequence (one per wave)
#define WPW 4      // waves per workgroup
#define WGSEQ (GROUPS / WPW) // 32 workgroups per sequence

typedef float v2f __attribute__((ext_vector_type(2)));
typedef float v8f __attribute__((ext_vector_type(8)));

__device__ __forceinline__ v8f zero8() {
    v8f z = {0.f, 0.f, 0.f, 0.f, 0.f, 0.f, 0.f, 0.f};
    return z;
}
__device__ __forceinline__ float red_max16(float x) {
#pragma unroll
    for (int off = 1; off < 16; off <<= 1) x = fmaxf(x, __shfl_xor(x, off, 32));
    return x;
}
__device__ __forceinline__ float red_sum16(float x) {
#pragma unroll
    for (int off = 1; off < 16; off <<= 1) x += __shfl_xor(x, off, 32);
    return x;
}

// ---------------------------------------------------------------------------
// Kernel 1: RoPE the new token's Q and K (llama half-dim convention).
// qr/kr: [B,H,D] in workspace.
// ---------------------------------------------------------------------------
__global__ void rope_kernel(const float* __restrict__ Q, const float* __restrict__ K,
                            const float* __restrict__ cosT, const float* __restrict__ sinT,
                            const int* __restrict__ cache_length,
                            float* __restrict__ qr, float* __restrict__ kr) {
    const int b = blockIdx.x;
    const int cl = cache_length[b];
    const float* crow = cosT + (size_t)cl * HD;
    const float* srow = sinT + (size_t)cl * HD;
    const size_t base = (size_t)b * NH * HD;
    for (int idx = threadIdx.x; idx < NH * HD; idx += blockDim.x) {
        const int i = idx & (HD - 1);
        const int hb = idx - i;
        const float cv = crow[i], sv = srow[i];
        const int ri = (i < HD / 2) ? (i + HD / 2) : (i - HD / 2);
        const float sgn = (i < HD / 2) ? -1.0f : 1.0f;
        qr[base + idx] = Q[base + idx] * cv + sgn * Q[base + hb + ri] * sv;
        kr[base + idx] = K[base + idx] * cv + sgn * K[base + hb + ri] * sv;
    }
}

// ---------------------------------------------------------------------------
// Kernel 2: flash-decode partials. One wave = one 16-key block, all 16 heads.
// Scores via chained V_WMMA_F32_16X16X4_F32 (A rows = broadcast q, so every
// accumulator row is the true score vector); PV via the same trick with p.
// Fully-masked blocks short-circuit: write (m=-inf, l=0), read no K/V.
// Position s == cache_length[b] overrides its K/V row with the new token.
// ---------------------------------------------------------------------------
__global__ __launch_bounds__(WPW * 32) void attn_partial_kernel(
    const float* __restrict__ Kc, const float* __restrict__ Vc,
    const float* __restrict__ Vnew,
    const float* __restrict__ qr, const float* __restrict__ kr,
    const float* __restrict__ mask,
    const int* __restrict__ cache_length, const int* __restrict__ fetch_slots,
    float* __restrict__ ml, float* __restrict__ outp) {
    const int b    = blockIdx.x / WGSEQ;
    const int wgs  = blockIdx.x % WGSEQ;
    const int wave = threadIdx.x >> 5;
    const int lane = threadIdx.x & 31;
    const int g    = wgs * WPW + wave;   // key-block index within sequence
    const int cl   = cache_length[b];
    const int sbase = g * BLK;
    const int rec0  = (b * GROUPS + g) * NH;

    if (sbase > cl) {  // wave-uniform: whole block past causal bound
        if (lane < NH) {
            ml[(size_t)(rec0 + lane) * 2 + 0] = -__builtin_inff();
            ml[(size_t)(rec0 + lane) * 2 + 1] = 0.0f;
        }
        return;
    }

    const int key = lane & 15;   // this lane's key column (B operand / C row 0)
    const int hi  = lane >> 4;   // which K-half of the WMMA this lane supplies
    const int s   = sbase + key;
    const bool valid = (s <= cl);
    const float mval = mask[(size_t)b * MAXS + s];
    const int blk = fetch_slots[b * BPS + g];
    const long slot = (long)blk * BLK;   // first flat cache row of this block
    const float scale = 0.088388347648318447f;  // 1/sqrt(128)

#pragma unroll 1
    for (int h = 0; h < NH; ++h) {
        const float* qrow = qr   + (size_t)(b * NH + h) * HD;
        const float* newK = kr   + (size_t)(b * NH + h) * HD;
        const float* newV = Vnew + (size_t)(b * NH + h) * HD;
        const float* krow = (s == cl) ? newK
                                      : (Kc + ((slot + key) * NH + h) * HD);

        // ---- scores: C = Q_bcast(16x128) x K^T(128x16), 32 chained WMMAs
        v8f c = zero8();
#pragma unroll
        for (int k0 = 0; k0 < HD; k0 += 4) {
            v2f a = *(const v2f*)(qrow + k0 + 2 * hi);  // A: broadcast rows
            v2f bm = *(const v2f*)(krow + k0 + 2 * hi); // B: K row per lane
            c = __builtin_amdgcn_wmma_f32_16x16x4_f32(
                    false, a, false, bm, (short)0, c, false, false);
        }
        float sc = valid ? (c[0] * scale + mval) : -__builtin_inff();
        float m = red_max16(sc);
        float p = valid ? __expf(sc - m) : 0.0f;
        float l = red_sum16(p);

        // ---- out tile: O(16x128) = P_bcast(16x16) x V(16x128)
        v8f o[8];
#pragma unroll
        for (int dt = 0; dt < 8; ++dt) o[dt] = zero8();
#pragma unroll
        for (int kc = 0; kc < BLK; kc += 4) {
            const int r0 = kc + 2 * hi;   // V rows this lane feeds (B layout)
            const int r1 = r0 + 1;
            v2f a;                         // A: p broadcast across rows
            a.x = __shfl(p, r0, 32);
            a.y = __shfl(p, r1, 32);
            const float* vr0 = (sbase + r0 == cl) ? newV
                               : (Vc + ((slot + r0) * NH + h) * HD);
            const float* vr1 = (sbase + r1 == cl) ? newV
                               : (Vc + ((slot + r1) * NH + h) * HD);
#pragma unroll
            for (int dt = 0; dt < 8; ++dt) {
                v2f bm;
                bm.x = vr0[dt * 16 + key];
                bm.y = vr1[dt * 16 + key];
                o[dt] = __builtin_amdgcn_wmma_f32_16x16x4_f32(
                            false, a, false, bm, (short)0, o[dt], false, false);
            }
        }

        if (lane == 0) {
            ml[(size_t)(rec0 + h) * 2 + 0] = m;
            ml[(size_t)(rec0 + h) * 2 + 1] = l;
        }
        if (lane < 16) {  // C row 0 lives in VGPR0 of lanes 0-15
            float* orow = outp + (size_t)(rec0 + h) * HD;
#pragma unroll
            for (int dt = 0; dt < 8; ++dt) orow[dt * 16 + lane] = o[dt][0];
        }
    }
}

// ---------------------------------------------------------------------------
// Kernel 3: combine the 128 partials per (b,h):
//   out = sum_g exp(m_g - M) * acc_g / sum_g exp(m_g - M) * l_g
// Skips never-written partials (m_g == -inf), block-uniform branch.
// ---------------------------------------------------------------------------
__global__ __launch_bounds__(HD) void attn_reduce_kernel(
    const float* __restrict__ ml, const float* __restrict__ outp,
    float* __restrict__ out) {
    const int bh = blockIdx.x;        // b*NH + h
    const int b = bh / NH, h = bh % NH;
    const int d = threadIdx.x;

    float M = -__builtin_inff();
    for (int g = 0; g < GROUPS; ++g)
        M = fmaxf(M, ml[(size_t)((b * GROUPS + g) * NH + h) * 2]);

    float L = 0.f, acc = 0.f;
    for (int g = 0; g < GROUPS; ++g) {
        const int rec = (b * GROUPS + g) * NH + h;
        const float mg = ml[(size_t)rec * 2];
        if (mg == -__builtin_inff()) continue;  // uniform across block
        const float w = __expf(mg - M);
        L += w * ml[(size_t)rec * 2 + 1];
        acc += w * outp[(size_t)rec * HD + d];
    }
    out[(size_t)bh * HD + d] = acc / L;
}

// ---------------------------------------------------------------------------
// Inputs (setup_inputs order): 0:Q 1:K 2:V 3:Kcache 4:Vcache 5:cos 6:sin
// 7:mask 8:input_length 9:cache_length 10:save_slots 11:fetch_slots 12:max_s
// ---------------------------------------------------------------------------
extern "C" void kernel_launch(void* const* d_in, const int* in_sizes, int n_in,
                              void* d_out, int out_size, void* d_ws, size_t ws_size,
                              hipStream_t stream) {
    const float* Q     = (const float*)d_in[0];
    const float* K     = (const float*)d_in[1];
    const float* V     = (const float*)d_in[2];
    const float* Kc    = (const float*)d_in[3];
    const float* Vc    = (const float*)d_in[4];
    const float* cosT  = (const float*)d_in[5];
    const float* sinT  = (const float*)d_in[6];
    const float* mask  = (const float*)d_in[7];
    const int* cache_length = (const int*)d_in[9];
    const int* fetch_slots  = (const int*)d_in[11];

    // Workspace layout (floats):
    //   qr [B*H*D] | kr [B*H*D] | ml [B*GROUPS*H*2] | outp [B*GROUPS*H*D]
    float* ws = (float*)d_ws;
    float* qr   = ws;
    float* kr   = qr + (size_t)NB * NH * HD;
    float* ml   = kr + (size_t)NB * NH * HD;
    float* outp = ml + (size_t)NB * GROUPS * NH * 2;

    rope_kernel<<<NB, 256, 0, stream>>>(Q, K, cosT, sinT, cache_length, qr, kr);
    attn_partial_kernel<<<NB * WGSEQ, WPW * 32, 0, stream>>>(
        Kc, Vc, V, qr, kr, mask, cache_length, fetch_slots, ml, outp);
    attn_reduce_kernel<<<NB * NH, HD, 0, stream>>>(ml, outp, (float*)d_out);
}